// MoERouter_23493471109262
// MI455X (gfx1250) — compile-verified
//
#include <hip/hip_runtime.h>
#include <hip/hip_bf16.h>

// MoE router for MI455X (gfx1250).
// GEMM: x[16384,4096] @ W^T[4096,64] via V_WMMA_F32_16X16X4_F32 (fp32 matrix path).
// W chunks staged into LDS with global_load_async_to_lds_b128 (ASYNCcnt);
// x streamed once from HBM (memory-bound, 256MB -> ~11us floor @ 23.3TB/s).

typedef __attribute__((ext_vector_type(2))) float v2f;
typedef __attribute__((ext_vector_type(4))) float v4f;
typedef __attribute__((ext_vector_type(8))) float v8f;

#define DIMK   4096      // hidden dim (K)
#define NEXP   64        // experts (N)
#define TOK    16384     // B*S tokens (M)
#define KC     128       // K-chunk staged in LDS
#define LSTR   132       // LDS row stride in dwords (132%64=4 -> conflict-free, 132*4%16==0 -> 16B-aligned)
#define NB2    128       // post-process blocks

// ---------------------------------------------------------------------------
// Kernel 1: router GEMM. grid=256 x 128 threads (4 waves). Each wave: 16 tokens
// x 64 experts, 4 WMMA accumulators. A from global (streamed), B from LDS.
// ---------------------------------------------------------------------------
__global__ __launch_bounds__(128) void router_gemm(const float* __restrict__ x,
                                                   const float* __restrict__ W,
                                                   float* __restrict__ logits) {
  __shared__ float ldsW[NEXP * LSTR];

  const int tid  = threadIdx.x;
  const int lane = tid & 31;
  const int wv   = tid >> 5;
  const int mloc = lane & 15;   // M (token) within 16, also N (expert) within 16
  const int half = lane >> 4;   // selects K pair (A/B frags) and M half (C/D)
  const int m0   = blockIdx.x * 64 + wv * 16;

  const float* __restrict__ xrow = x + (size_t)(m0 + mloc) * DIMK;

  v8f acc0 = {}, acc1 = {}, acc2 = {}, acc3 = {};

  for (int kc = 0; kc < DIMK; kc += KC) {
    __syncthreads();
    // Async cooperative fill: W[0:64][kc:kc+KC] -> LDS (row-major, padded stride).
    // LDS dest address = low 32 bits of flat shared-aperture address.
    #pragma unroll
    for (int i = tid; i < NEXP * (KC / 4); i += 128) {
      const int row = i >> 5;        // KC/4 == 32 float4 per row
      const int c4  = i & 31;
      const unsigned ldsAddr = (unsigned)(size_t)&ldsW[row * LSTR + c4 * 4];
      const unsigned gOff    = (unsigned)(((size_t)row * DIMK + kc + c4 * 4) * sizeof(float));
      asm volatile("global_load_async_to_lds_b128 %0, %1, %2"
                   :: "v"(ldsAddr), "v"(gOff), "s"(W)
                   : "memory");
    }
    asm volatile("s_wait_asynccnt 0x0" ::: "memory");
    __syncthreads();

    if (kc + KC < DIMK) {
      __builtin_prefetch(xrow + kc + KC, 0, 0);  // global_prefetch_b8, next x chunk
    }

    #pragma unroll 4
    for (int kk = 0; kk < KC; kk += 4) {
      // A frag (16x4 fp32): lane holds x[m][k+2*half], x[m][k+2*half+1]
      v2f a = *(const v2f*)(xrow + kc + kk + 2 * half);
      const int bo = kk + 2 * half;
      // B frags (4x16 fp32): lane holds W[n][k+2*half], W[n][k+2*half+1]
      v2f b0 = *(const v2f*)&ldsW[(mloc +  0) * LSTR + bo];
      v2f b1 = *(const v2f*)&ldsW[(mloc + 16) * LSTR + bo];
      v2f b2 = *(const v2f*)&ldsW[(mloc + 32) * LSTR + bo];
      v2f b3 = *(const v2f*)&ldsW[(mloc + 48) * LSTR + bo];
      acc0 = __builtin_amdgcn_wmma_f32_16x16x4_f32(false, a, false, b0, (short)0, acc0, false, false);
      acc1 = __builtin_amdgcn_wmma_f32_16x16x4_f32(false, a, false, b1, (short)0, acc1, false, false);
      acc2 = __builtin_amdgcn_wmma_f32_16x16x4_f32(false, a, false, b2, (short)0, acc2, false, false);
      acc3 = __builtin_amdgcn_wmma_f32_16x16x4_f32(false, a, false, b3, (short)0, acc3, false, false);
    }
  }

  // C/D layout: VGPR r, lanes 0-15 -> M=r, lanes 16-31 -> M=r+8; N = lane%16.
  #pragma unroll
  for (int r = 0; r < 8; ++r) {
    const int token = m0 + r + 8 * half;
    float* dst = logits + (size_t)token * NEXP + mloc;
    dst[0]  = acc0[r];
    dst[16] = acc1[r];
    dst[32] = acc2[r];
    dst[48] = acc3[r];
  }
}

// ---------------------------------------------------------------------------
// Kernel 2: per-token post-processing + deterministic block partial reductions.
// grid = NB2 x 128 threads, one token per thread.
// ---------------------------------------------------------------------------
__global__ __launch_bounds__(128) void router_post(const float* __restrict__ logits,
                                                   float* __restrict__ out_ew,
                                                   float* __restrict__ out_ei,
                                                   float* __restrict__ out_pr,
                                                   float* __restrict__ usagePartial,
                                                   float* __restrict__ zPartial) {
  __shared__ float tile[128 * 65];   // transpose tile for deterministic usage sum
  __shared__ float zred[128];

  const int tid = threadIdx.x;
  const int g   = blockIdx.x * 128 + tid;   // token index
  const float* __restrict__ lg = logits + (size_t)g * NEXP;

  float l[NEXP];
  #pragma unroll
  for (int i = 0; i < 16; ++i) {
    v4f v = *(const v4f*)(lg + i * 4);
    l[i * 4 + 0] = v[0];
    l[i * 4 + 1] = v[1];
    l[i * 4 + 2] = v[2];
    l[i * 4 + 3] = v[3];
  }

  // max / sum-exp / logsumexp
  float mx = l[0];
  #pragma unroll
  for (int e = 1; e < NEXP; ++e) mx = fmaxf(mx, l[e]);
  float sum = 0.0f;
  #pragma unroll
  for (int e = 0; e < NEXP; ++e) sum += __expf(l[e] - mx);
  const float lse = mx + __logf(sum);

  // top-2 (earliest index wins on ties, matching lax.top_k)
  float v0 = -3.402823466e38f, v1 = -3.402823466e38f;
  int   i0 = 0, i1 = 0;
  #pragma unroll
  for (int e = 0; e < NEXP; ++e) {
    const float le = l[e];
    if (le > v0) { v1 = v0; i1 = i0; v0 = le; i0 = e; }
    else if (le > v1) { v1 = le; i1 = e; }
  }
  const float e1 = __expf(v1 - v0);
  const float dn = 1.0f / (1.0f + e1);
  out_ew[(size_t)g * 2 + 0] = dn;
  out_ew[(size_t)g * 2 + 1] = e1 * dn;
  out_ei[(size_t)g * 2 + 0] = (float)i0;
  out_ei[(size_t)g * 2 + 1] = (float)i1;

  // full softmax probs
  const float inv = 1.0f / sum;
  float p[NEXP];
  #pragma unroll
  for (int e = 0; e < NEXP; ++e) p[e] = __expf(l[e] - mx) * inv;

  float* __restrict__ pr = out_pr + (size_t)g * NEXP;
  #pragma unroll
  for (int i = 0; i < 16; ++i) {
    v4f v;
    v[0] = p[i * 4 + 0]; v[1] = p[i * 4 + 1];
    v[2] = p[i * 4 + 2]; v[3] = p[i * 4 + 3];
    *(v4f*)(pr + i * 4) = v;
  }

  // deterministic reductions: transpose through LDS, fixed summation order
  #pragma unroll
  for (int e = 0; e < NEXP; ++e) tile[tid * 65 + e] = p[e];
  zred[tid] = lse * lse;
  __syncthreads();

  if (tid < NEXP) {
    float s = 0.0f;
    for (int t = 0; t < 128; ++t) s += tile[t * 65 + tid];
    usagePartial[(size_t)blockIdx.x * NEXP + tid] = s;
  }
  if (tid == 0) {
    float zs = 0.0f;
    for (int t = 0; t < 128; ++t) zs += zred[t];
    zPartial[blockIdx.x] = zs;
  }
}

// ---------------------------------------------------------------------------
// Kernel 3: fold partials -> expert_usage, load_balance_loss, router_z_loss.
// ---------------------------------------------------------------------------
__global__ __launch_bounds__(64) void router_final(const float* __restrict__ usagePartial,
                                                   const float* __restrict__ zPartial,
                                                   float* __restrict__ out_lb,
                                                   float* __restrict__ out_z,
                                                   float* __restrict__ out_usage) {
  __shared__ float usq[NEXP];
  const int e = threadIdx.x;
  float s = 0.0f;
  for (int b = 0; b < NB2; ++b) s += usagePartial[(size_t)b * NEXP + e];
  const float u = s * (1.0f / (float)TOK);
  out_usage[e] = u;
  usq[e] = u * u;
  __syncthreads();
  if (e == 0) {
    float lb = 0.0f;
    for (int i = 0; i < NEXP; ++i) lb += usq[i];
    out_lb[0] = lb * (float)NEXP;
    float zs = 0.0f;
    for (int b = 0; b < NB2; ++b) zs += zPartial[b];
    out_z[0] = zs * (1.0f / (float)TOK);
  }
}

// ---------------------------------------------------------------------------
extern "C" void kernel_launch(void* const* d_in, const int* in_sizes, int n_in,
                              void* d_out, int out_size, void* d_ws, size_t ws_size,
                              hipStream_t stream) {
  const float* x = (const float*)d_in[0];   // [4,4096,4096] fp32
  const float* W = (const float*)d_in[1];   // [64,4096] fp32
  float* out = (float*)d_out;

  // d_out layout (flat, return order):
  float* out_ew = out;                 // expert_weights  [16384,2]
  float* out_ei = out + 32768;         // expert_indices  [16384,2] (as float)
  float* out_lb = out + 65536;         // load_balance_loss scalar
  float* out_z  = out + 65537;         // router_z_loss scalar
  float* out_us = out + 65538;         // expert_usage    [64]
  float* out_pr = out + 65602;         // routing_probs   [16384,64]

  // workspace: logits + partials
  float* logits = (float*)d_ws;                         // 16384*64
  float* usageP = logits + (size_t)TOK * NEXP;          // NB2*64
  float* zP     = usageP + (size_t)NB2 * NEXP;          // NB2

  router_gemm <<<TOK / 64, 128, 0, stream>>>(x, W, logits);
  router_post <<<TOK / 128, 128, 0, stream>>>(logits, out_ew, out_ei, out_pr, usageP, zP);
  router_final<<<1, 64, 0, stream>>>(usageP, zP, out_lb, out_z, out_us);
}